// GraphConv_9706626090092
// MI455X (gfx1250) — compile-verified
//
#include <hip/hip_runtime.h>

#define N_NODES   100000
#define N_EDGES   1600000
#define IN_FEATS  128
#define OUT_FEATS 64
#define NODE_PAD  102400   // padded float count for norm region in ws
#define N_STRIPS  (N_NODES / 16)   // 6250, exact

typedef __attribute__((ext_vector_type(2))) float v2f;
typedef __attribute__((ext_vector_type(8))) float v8f;

// ---------------------------------------------------------------- utilities
__global__ __launch_bounds__(256) void k_zero(float* __restrict__ p, int n) {
  int i = blockIdx.x * 256 + threadIdx.x;
  if (i < n) p[i] = 0.0f;
}

// degree histogram: exact + deterministic (sum of equal 1.0f values)
__global__ __launch_bounds__(256) void k_degree(const int* __restrict__ src,
                                                float* __restrict__ degs) {
  int e = blockIdx.x * 256 + threadIdx.x;
  if (e < N_EDGES) atomicAdd(&degs[src[e]], 1.0f);
}

__global__ __launch_bounds__(256) void k_norm(float* __restrict__ norm) {
  int i = blockIdx.x * 256 + threadIdx.x;
  if (i < N_NODES) {
    float d = norm[i];
    d = d < 1.0f ? 1.0f : d;
    norm[i] = 1.0f / sqrtf(d);   // refined sequence, not raw v_rsq approx
  }
}

// ------------------------------------------------- WMMA GEMM: h = feat@W * norm
// One wave32 per 16-row strip. 4 accumulators tile N=64; K=128 in steps of 4.
__global__ __launch_bounds__(256) void k_gemm(const float* __restrict__ feat,
                                              const float* __restrict__ W,
                                              const float* __restrict__ norm,
                                              float* __restrict__ h) {
  const int strip = (int)((blockIdx.x * 256u + threadIdx.x) >> 5);
  if (strip >= N_STRIPS) return;           // whole-wave uniform branch
  const int lane = threadIdx.x & 31;
  const int half = lane >> 4;              // 0: lanes 0-15, 1: lanes 16-31
  const int lr   = lane & 15;
  const int m0   = strip * 16;

  v8f acc0 = {}, acc1 = {}, acc2 = {}, acc3 = {};

  // A 16x4 f32 layout: lane<16 -> K={k0,k0+1}; lane>=16 -> K={k0+2,k0+3}
  const float* arow = feat + (size_t)(m0 + lr) * IN_FEATS + 2 * half;
  const float* wcol = W + lr;

#pragma unroll 4
  for (int k0 = 0; k0 < IN_FEATS; k0 += 4) {
    v2f a = { arow[k0], arow[k0 + 1] };

    // B 4x16 f32 layout mirrors A across lane halves
    const float* wr0 = wcol + (size_t)(k0 + 2 * half) * OUT_FEATS;
    const float* wr1 = wr0 + OUT_FEATS;
    v2f b0 = { wr0[0],  wr1[0]  };
    v2f b1 = { wr0[16], wr1[16] };
    v2f b2 = { wr0[32], wr1[32] };
    v2f b3 = { wr0[48], wr1[48] };

    acc0 = __builtin_amdgcn_wmma_f32_16x16x4_f32(false, a, false, b0, (short)0, acc0, false, false);
    acc1 = __builtin_amdgcn_wmma_f32_16x16x4_f32(false, a, false, b1, (short)0, acc1, false, false);
    acc2 = __builtin_amdgcn_wmma_f32_16x16x4_f32(false, a, false, b2, (short)0, acc2, false, false);
    acc3 = __builtin_amdgcn_wmma_f32_16x16x4_f32(false, a, false, b3, (short)0, acc3, false, false);
  }

  // C/D 16x16: VGPR g -> row m0 + g + 8*half, col = lr (+ n-tile offset)
#pragma unroll
  for (int g = 0; g < 8; ++g) {
    const int row = m0 + g + 8 * half;
    const float s = norm[row];
    float* orow = h + (size_t)row * OUT_FEATS + lr;
    orow[0]  = acc0[g] * s;
    orow[16] = acc1[g] * s;
    orow[32] = acc2[g] * s;
    orow[48] = acc3[g] * s;
  }
}

// ----------------------------------------------- edge scatter: out[dst] += h[src]
// 64 contiguous lanes per edge: coalesced h reads (L2-resident), contiguous
// 256B atomic bursts per destination row (atomics resolve in 192MB L2).
__global__ __launch_bounds__(256) void k_scatter(const int* __restrict__ src,
                                                 const int* __restrict__ dst,
                                                 const float* __restrict__ h,
                                                 float* __restrict__ out) {
  long long i = (long long)blockIdx.x * 256 + threadIdx.x;
  if (i >= (long long)N_EDGES * OUT_FEATS) return;
  const int e = (int)(i >> 6);
  const int j = (int)(i & 63);
  atomicAdd(&out[(size_t)dst[e] * OUT_FEATS + j],
            h[(size_t)src[e] * OUT_FEATS + j]);
}

// ------------------------------------------------------- post-scale + bias
__global__ __launch_bounds__(256) void k_finalize(float* __restrict__ out,
                                                  const float* __restrict__ norm,
                                                  const float* __restrict__ bias) {
  int i = blockIdx.x * 256 + threadIdx.x;
  if (i < N_NODES * OUT_FEATS) {
    out[i] = out[i] * norm[i >> 6] + bias[i & 63];
  }
}

extern "C" void kernel_launch(void* const* d_in, const int* in_sizes, int n_in,
                              void* d_out, int out_size, void* d_ws, size_t ws_size,
                              hipStream_t stream) {
  const float* feat = (const float*)d_in[0];
  const float* W    = (const float*)d_in[1];
  const float* bias = (const float*)d_in[2];
  const int*   src  = (const int*)d_in[3];
  const int*   dst  = (const int*)d_in[4];
  float* out  = (float*)d_out;
  float* norm = (float*)d_ws;               // N_NODES floats (degs then norm)
  float* h    = (float*)d_ws + NODE_PAD;    // N_NODES * OUT_FEATS floats

  k_zero<<<(N_NODES + 255) / 256, 256, 0, stream>>>(norm, N_NODES);
  k_zero<<<(N_NODES * OUT_FEATS + 255) / 256, 256, 0, stream>>>(out, N_NODES * OUT_FEATS);
  k_degree<<<(N_EDGES + 255) / 256, 256, 0, stream>>>(src, norm);
  k_norm<<<(N_NODES + 255) / 256, 256, 0, stream>>>(norm);
  k_gemm<<<(N_STRIPS + 7) / 8, 256, 0, stream>>>(feat, W, norm, h);  // 8 waves/block
  const long long tot = (long long)N_EDGES * OUT_FEATS;
  k_scatter<<<(unsigned)((tot + 255) / 256), 256, 0, stream>>>(src, dst, h, out);
  k_finalize<<<(N_NODES * OUT_FEATS + 255) / 256, 256, 0, stream>>>(out, norm, bias);
}